// SuperVoxel_12953621365098
// MI455X (gfx1250) — compile-verified
//
#include <hip/hip_runtime.h>
#include <math.h>

typedef __attribute__((ext_vector_type(2))) float v2f;
typedef __attribute__((ext_vector_type(8))) float v8f;

#define TPB 256
#define ITER 8
#define IMG_W 1024
#define IMG_H 1024

// ---------------- union-find (Komura/Playne lock-free, atomicMin-based) ----------

__device__ __forceinline__ int find_root_mut(int* L, int x) {
    // chase with atomicMin path-halving; all writes monotonically decrease,
    // parent[x] <= x invariant guarantees termination.
    while (true) {
        int p = L[x];
        if (p == x) return x;
        int gp = L[p];
        if (gp == p) return p;
        atomicMin(&L[x], gp);
        x = gp;
    }
}

__device__ __forceinline__ int find_root_ro(const int* L, int x) {
    int p = L[x];
    while (p != x) { x = p; p = L[x]; }
    return x;
}

__device__ __forceinline__ void unite(int* L, int a, int b) {
    a = find_root_mut(L, a);
    b = find_root_mut(L, b);
    while (a != b) {
        if (a < b) {
            int old = atomicMin(&L[b], a);
            if (old == b) break;
            b = old;
        } else {
            int old = atomicMin(&L[a], b);
            if (old == a) break;
            a = old;
        }
    }
}

__device__ __forceinline__ unsigned getbit(const unsigned* w, int i) {
    return (w[i >> 5] >> (i & 31)) & 1u;
}

// ---------------- kernel 1: init parents, fg bitmasks (wave32 ballot), zero flags

__global__ void k_init(const float* __restrict__ preds, const float* __restrict__ targets,
                       int* parentP, int* parentT,
                       unsigned* fgP, unsigned* fgT,
                       unsigned* flagPos, unsigned* flagNeg,
                       int N, int fgWords) {
    int i = blockIdx.x * TPB + threadIdx.x;
    bool inr = i < N;
    int ic = inr ? i : (N - 1);
    if (inr) { parentP[i] = i; parentT[i] = i; }
    bool pf = inr && (preds[ic] > 0.5f);
    bool tf = inr && (targets[ic] > 0.0f);
    unsigned long long mp = __ballot(pf);
    unsigned long long mt = __ballot(tf);
    int word = i >> 5;
    if ((threadIdx.x & 31) == 0 && word < fgWords) {
        fgP[word] = (unsigned)mp;
        fgT[word] = (unsigned)mt;
    }
    if (i < fgWords) { flagPos[i] = 0u; flagNeg[i] = 0u; }
}

// ---------------- kernel 2: 4-connectivity unions (right + down edges, per image)

__global__ void k_union(const unsigned* __restrict__ fgP, const unsigned* __restrict__ fgT,
                        int* parentP, int* parentT, int N) {
    int i = blockIdx.x * TPB + threadIdx.x;
    if (i >= N) return;
    int x = i & (IMG_W - 1);
    int yy = (i / IMG_W) & (IMG_H - 1);   // row within image
    bool bp = getbit(fgP, i), bt = getbit(fgT, i);
    if (x < IMG_W - 1) {
        if (bp && getbit(fgP, i + 1)) unite(parentP, i, i + 1);
        if (bt && getbit(fgT, i + 1)) unite(parentT, i, i + 1);
    }
    if (yy < IMG_H - 1) {
        if (bp && getbit(fgP, i + IMG_W)) unite(parentP, i, i + IMG_W);
        if (bt && getbit(fgT, i + IMG_W)) unite(parentT, i, i + IMG_W);
    }
}

// ---------------- kernel 3: flatten (read-only find, then store final root)

__global__ void k_flatten(int* parentP, int* parentT, int N) {
    int i = blockIdx.x * TPB + threadIdx.x;
    if (i >= N) return;
    parentP[i] = find_root_ro(parentP, i);
    parentT[i] = find_root_ro(parentT, i);
}

// ---------------- kernel 4: per-component "has a mistake" flags (idempotent atomicOr)

__global__ void k_flags(const unsigned* __restrict__ fgP, const unsigned* __restrict__ fgT,
                        const int* __restrict__ parentP, const int* __restrict__ parentT,
                        unsigned* flagPos, unsigned* flagNeg, int N) {
    int i = blockIdx.x * TPB + threadIdx.x;
    if (i >= N) return;
    bool bp = getbit(fgP, i), bt = getbit(fgT, i);
    if (bp && !bt) {            // pred component contains a false-positive pixel
        int r = parentP[i];
        atomicOr(&flagPos[r >> 5], 1u << (r & 31));
    }
    if (bt && !bp) {            // target component contains a missed pixel
        int r = parentT[i];
        atomicOr(&flagNeg[r >> 5], 1u << (r & 31));
    }
}

// ---------------- kernel 5: fused BCE loss + critical weight, WMMA fp32 accumulation
// float4/int4 vectorized: 4 pixels per thread per iteration, fully branchless body
// (EXEC statically all-1s as WMMA requires), one v_wmma per 4 pixels, unrolled x8.

__global__ void k_loss(const float4* __restrict__ preds4, const float4* __restrict__ targets4,
                       const unsigned* __restrict__ fgP, const unsigned* __restrict__ fgT,
                       const int4* __restrict__ parentP4, const int4* __restrict__ parentT4,
                       const unsigned* __restrict__ flagPos, const unsigned* __restrict__ flagNeg,
                       float* partials, int N4, int N) {
    const int total = gridDim.x * TPB;
    const int base = blockIdx.x * TPB + threadIdx.x;

    v8f acc = {};                          // 16x16 f32 accumulator in the matrix pipe
    const v2f a_ones = {1.0f, 1.0f};       // A = ones(16x4): D[m][n] = sum_k B[k][n] + C

    #pragma unroll
    for (int k = 0; k < ITER; ++k) {
        int g = base + k * total;
        int gc = g < N4 ? g : (N4 - 1);    // clamp; keep EXEC all-1s for WMMA
        int pg = gc + total; pg = pg < N4 ? pg : (N4 - 1);
        __builtin_prefetch(&preds4[pg], 0, 0);     // global_prefetch_b8
        __builtin_prefetch(&targets4[pg], 0, 0);

        float4 p4 = preds4[gc];
        float4 t4 = targets4[gc];
        int4 rP4 = parentP4[gc];
        int4 rT4 = parentT4[gc];

        int i0 = gc << 2;                   // first pixel of the group (4-aligned)
        unsigned wP = fgP[i0 >> 5] >> (i0 & 31);   // 4 consecutive fg bits, one word
        unsigned wT = fgT[i0 >> 5] >> (i0 & 31);

        float v[4];
        const float pe[4] = {p4.x, p4.y, p4.z, p4.w};
        const float te[4] = {t4.x, t4.y, t4.z, t4.w};
        const int rpe[4] = {rP4.x, rP4.y, rP4.z, rP4.w};
        const int rte[4] = {rT4.x, rT4.y, rT4.z, rT4.w};
        #pragma unroll
        for (int e = 0; e < 4; ++e) {
            float p = pe[e], t = te[e];
            float loss = fmaxf(p, 0.0f) - p * t + log1pf(expf(-fabsf(p)));
            int rP = rpe[e], rT = rte[e];
            // branchless: flag arrays are tiny (L1/L2 resident) -> always load
            unsigned posb = ((wP >> e) & 1u) & ((flagPos[rP >> 5] >> (rP & 31)) & 1u);
            unsigned negb = ((wT >> e) & 1u) & ((flagNeg[rT >> 5] >> (rT & 31)) & 1u);
            float w = 0.5f + 0.25f * (float)negb + 0.25f * (float)posb;
            float valid = ((g << 2) + e < N) ? 1.0f : 0.0f;
            v[e] = valid * w * loss;
        }

        v2f bvec = {v[0] + v[2], v[1] + v[3]};
        // D = ones(16x4) x B(4x16) + C  -> every row of D holds the column sums
        acc = __builtin_amdgcn_wmma_f32_16x16x4_f32(
            false, a_ones, false, bvec, (short)0, acc, false, false);
    }

    // acc[0] lanes 0..15: row M=0 column sums; lanes 16..31: row M=8 (identical).
    float s = acc[0];
    for (int off = 16; off > 0; off >>= 1) s += __shfl_xor(s, off, 32);
    s *= 0.5f;                              // rows duplicated -> halve

    __shared__ float wsum[TPB / 32];
    int wave = threadIdx.x >> 5;
    if ((threadIdx.x & 31) == 0) wsum[wave] = s;
    __syncthreads();
    if (threadIdx.x == 0) {
        float tsum = 0.0f;
        for (int w2 = 0; w2 < TPB / 32; ++w2) tsum += wsum[w2];
        partials[blockIdx.x] = tsum;
    }
}

// ---------------- kernel 6: deterministic final reduction, mean

__global__ void k_final(const float* __restrict__ partials, int nb, float* out, int N) {
    __shared__ float sm[256];
    float s = 0.0f;
    for (int i = threadIdx.x; i < nb; i += 256) s += partials[i];
    sm[threadIdx.x] = s;
    __syncthreads();
    for (int off = 128; off > 0; off >>= 1) {
        if (threadIdx.x < off) sm[threadIdx.x] += sm[threadIdx.x + off];
        __syncthreads();
    }
    if (threadIdx.x == 0) out[0] = sm[0] / (float)N;
}

// ---------------- launcher --------------------------------------------------------

extern "C" void kernel_launch(void* const* d_in, const int* in_sizes, int n_in,
                              void* d_out, int out_size, void* d_ws, size_t ws_size,
                              hipStream_t stream) {
    const float* preds = (const float*)d_in[0];
    const float* targets = (const float*)d_in[1];
    float* out = (float*)d_out;
    const int N = in_sizes[0];
    const int N4 = (N + 3) / 4;
    const int fgWords = (N + 31) / 32;

    char* ws = (char*)d_ws;
    int* parentP = (int*)ws;           ws += (size_t)N * sizeof(int);
    int* parentT = (int*)ws;           ws += (size_t)N * sizeof(int);
    unsigned* fgP = (unsigned*)ws;     ws += (size_t)fgWords * sizeof(unsigned);
    unsigned* fgT = (unsigned*)ws;     ws += (size_t)fgWords * sizeof(unsigned);
    unsigned* flagPos = (unsigned*)ws; ws += (size_t)fgWords * sizeof(unsigned);
    unsigned* flagNeg = (unsigned*)ws; ws += (size_t)fgWords * sizeof(unsigned);

    const int blocksN = (N + TPB - 1) / TPB;
    const int lossBlocks = (N4 + TPB * ITER - 1) / (TPB * ITER);
    float* partials = (float*)ws;      ws += (size_t)lossBlocks * sizeof(float);

    k_init<<<blocksN, TPB, 0, stream>>>(preds, targets, parentP, parentT,
                                        fgP, fgT, flagPos, flagNeg, N, fgWords);
    k_union<<<blocksN, TPB, 0, stream>>>(fgP, fgT, parentP, parentT, N);
    k_flatten<<<blocksN, TPB, 0, stream>>>(parentP, parentT, N);
    k_flags<<<blocksN, TPB, 0, stream>>>(fgP, fgT, parentP, parentT,
                                         flagPos, flagNeg, N);
    k_loss<<<lossBlocks, TPB, 0, stream>>>((const float4*)preds, (const float4*)targets,
                                           fgP, fgT,
                                           (const int4*)parentP, (const int4*)parentT,
                                           flagPos, flagNeg, partials, N4, N);
    k_final<<<1, 256, 0, stream>>>(partials, lossBlocks, out, N);
}